// FocalLoss_50173807951990
// MI455X (gfx1250) — compile-verified
//
#include <hip/hip_runtime.h>

// Problem shape (fixed by the reference).
#define BB 8
#define AA 100000
#define CC 80
#define MM 64
#define K1T 256
#define NBLK1 ((AA + K1T - 1) / K1T)      // 391 blocks per batch for assignment
#define K2BPB 512                         // blocks per batch for cls-loss stream
#define NQ ((AA * CC) / 4)                // float4 chunks per batch = 2,000,000

typedef __attribute__((ext_vector_type(2))) float v2f;
typedef __attribute__((ext_vector_type(4))) float v4f;
typedef __attribute__((ext_vector_type(8))) float v8f;

// Full-wave (32 lane) sum using V_WMMA_F32_16X16X4_F32 against an all-ones B.
// A layout (ISA 7.12.2, 32-bit A 16x4): lanes 0-15 hold K=0 (VGPR0), K=1 (VGPR1);
// lanes 16-31 hold K=2,3. With a = {s,0}: A[m,0]=s_m, A[m,2]=s_{m+16}.
// D[m,n] = s_m + s_{m+16}. Lane L<16 holds rows 0..7 in its 8 D regs, lanes>=16
// rows 8..15 -> per-lane sum of D regs + shfl_xor(16) = full wave sum.
// Call only at full-EXEC reconvergence points (WMMA requires EXEC all ones).
__device__ __forceinline__ float wave_sum32(float v) {
  v2f a; a.x = v;    a.y = 0.0f;
  v2f b; b.x = 1.0f; b.y = 1.0f;
  v8f c = {0.f, 0.f, 0.f, 0.f, 0.f, 0.f, 0.f, 0.f};
  v8f d = __builtin_amdgcn_wmma_f32_16x16x4_f32(false, a, false, b,
                                                (short)0, c, false, false);
  float r = ((d[0] + d[1]) + (d[2] + d[3])) + ((d[4] + d[5]) + (d[6] + d[7]));
  r += __shfl_xor(r, 16, 32);
  return r;
}

// ---------------------------------------------------------------------------
// Kernel 1: per-anchor assignment (IoU argmax), per-anchor code, and per-block
// partial num_pos / reg_sum. Grid: (NBLK1, B), block 256.
// ---------------------------------------------------------------------------
__global__ __launch_bounds__(256) void assign_kernel(
    const float* __restrict__ anchors,   // (A,4)
    const float* __restrict__ ann,       // (B,M,5)
    const float* __restrict__ regr,      // (B,A,4)
    int*   __restrict__ code,            // (B,A): >=0 pos class, -2 neg, -3 ignore
    float* __restrict__ nposp,           // (B,NBLK1)
    float* __restrict__ regp)            // (B,NBLK1)
{
  __shared__ float sX0[MM], sY0[MM], sX1[MM], sY1[MM], sAr[MM], sLb[MM];
  const int tid = threadIdx.x;
  const int b   = blockIdx.y;

  if (tid < MM) {
    const float* ap = ann + ((size_t)b * MM + tid) * 5;
    float x0 = ap[0], y0 = ap[1], x1 = ap[2], y1 = ap[3], lb = ap[4];
    float ar = (x1 - x0) * (y1 - y0);
    if (lb == -1.0f) {  // degenerate box: IoU becomes exactly 0, never wins a tie > 0
      x0 = 3.0e8f; y0 = 3.0e8f; x1 = -3.0e8f; y1 = -3.0e8f; ar = 0.0f;
    }
    sX0[tid] = x0; sY0[tid] = y0; sX1[tid] = x1; sY1[tid] = y1;
    sAr[tid] = ar; sLb[tid] = lb;
  }
  __syncthreads();

  const int a     = blockIdx.x * K1T + tid;
  const bool live = (a < AA);
  const int ac    = live ? a : (AA - 1);
  const v4f av = ((const v4f*)anchors)[ac];
  const float ax0 = av.x, ay0 = av.y, ax1 = av.z, ay1 = av.w;
  const float areaA = (ax1 - ax0) * (ay1 - ay0);

  float best = -1.0e30f; int bm = 0;
#pragma unroll 8
  for (int m = 0; m < MM; ++m) {
    float iw = fminf(ax1, sX1[m]) - fmaxf(ax0, sX0[m]);
    float ih = fminf(ay1, sY1[m]) - fmaxf(ay0, sY0[m]);
    float inter = fmaxf(iw, 0.0f) * fmaxf(ih, 0.0f);
    float ua = fmaxf(areaA + sAr[m] - inter, 1.0e-8f);
    float iou = inter * __builtin_amdgcn_rcpf(ua);
    if (iou > best) { best = iou; bm = m; }   // strict '>' => first-occurrence argmax
  }

  const bool pos = live && (best > 0.1f);
  const bool neg = (best <= 0.01f);
  float pf = pos ? 1.0f : 0.0f;
  float rs = 0.0f;

  if (pos) {
    float aw = ax1 - ax0, ah = ay1 - ay0;
    float acx = ax0 + 0.5f * aw, acy = ay0 + 0.5f * ah;
    float gx0 = sX0[bm], gy0 = sY0[bm], gx1 = sX1[bm], gy1 = sY1[bm];
    float gw0 = gx1 - gx0, gh0 = gy1 - gy0;
    float gcx = gx0 + 0.5f * gw0, gcy = gy0 + 0.5f * gh0;
    float gw = fmaxf(gw0, 1.0f), gh = fmaxf(gh0, 1.0f);
    float raw = __builtin_amdgcn_rcpf(aw), rah = __builtin_amdgcn_rcpf(ah);
    float t0 = (gcx - acx) * raw * 10.0f;
    float t1 = (gcy - acy) * rah * 10.0f;
    float t2 = __logf(gw * raw) * 5.0f;
    float t3 = __logf(gh * rah) * 5.0f;
    v4f rv = ((const v4f*)regr)[(size_t)b * AA + a];
    float d0 = fabsf(t0 - rv.x), d1 = fabsf(t1 - rv.y);
    float d2 = fabsf(t2 - rv.z), d3 = fabsf(t3 - rv.w);
    const float beta = 1.0f / 9.0f, hb = 0.5f / 9.0f;
    rs  = (d0 < beta) ? 4.5f * d0 * d0 : d0 - hb;
    rs += (d1 < beta) ? 4.5f * d1 * d1 : d1 - hb;
    rs += (d2 < beta) ? 4.5f * d2 * d2 : d2 - hb;
    rs += (d3 < beta) ? 4.5f * d3 * d3 : d3 - hb;
  }

  if (live) {
    int cidx = (int)fmaxf(sLb[bm], 0.0f);
    code[(size_t)b * AA + a] = pos ? cidx : (neg ? -2 : -3);
  }

  // Full-EXEC reconvergence point: WMMA wave reductions.
  float tp = wave_sum32(pf);
  float tr = wave_sum32(rs);

  __shared__ float sWp[8], sWr[8];
  if ((tid & 31) == 0) { sWp[tid >> 5] = tp; sWr[tid >> 5] = tr; }
  __syncthreads();
  if (tid == 0) {
    float sp = 0.f, sr = 0.f;
    for (int w = 0; w < 8; ++w) { sp += sWp[w]; sr += sWr[w]; }
    nposp[b * NBLK1 + blockIdx.x] = sp;
    regp [b * NBLK1 + blockIdx.x] = sr;
  }
}

// ---------------------------------------------------------------------------
// Kernel 2: focal classification loss over (B,A,C). Streams 256 MB with NT
// b128 loads + prefetch. Grid: (K2BPB, B), block 256. 4 | C so a float4 never
// straddles an anchor row; grid.y = batch so blocks never straddle batches.
// ---------------------------------------------------------------------------
__global__ __launch_bounds__(256) void clsloss_kernel(
    const float* __restrict__ cls,       // (B,A,C)
    const int*   __restrict__ code,      // (B,A)
    float* __restrict__ clsp)            // (B,K2BPB)
{
  const int tid = threadIdx.x;
  const int b   = blockIdx.y;
  const v4f* p4 = (const v4f*)(cls + (size_t)b * AA * CC);
  const int* cbase = code + (size_t)b * AA;
  const int step = K2BPB * 256;

  float acc = 0.0f;
  for (int q = blockIdx.x * 256 + tid; q < NQ; q += step) {
    __builtin_prefetch(p4 + q + step, 0, 0);         // global_prefetch_b8
    v4f x = __builtin_nontemporal_load(p4 + q);      // read-once stream: TH=NT
    int e  = q << 2;
    int an = e / CC;                                 // const divisor -> mul/shift
    int c0 = e - an * CC;
    int cd = cbase[an];
    float l = 0.0f;
#pragma unroll
    for (int j = 0; j < 4; ++j) {
      float xv = (j == 0) ? x.x : (j == 1) ? x.y : (j == 2) ? x.z : x.w;
      float p = fminf(fmaxf(xv, 1.0e-4f), 1.0f - 1.0e-4f);
      bool isone = (cd == c0 + j);                   // only possible when pos
      float qv = isone ? (1.0f - p) : p;
      float w  = isone ? 0.25f : 0.75f;
      // focal * bce = w * q^2 * (-ln(1-q))  [covers both target=1 and target=0]
      l += w * qv * qv * (-__logf(1.0f - qv));
    }
    acc += (cd == -3) ? 0.0f : l;                    // ignore anchors contribute 0
  }

  float t = wave_sum32(acc);                         // full-EXEC after loop exit
  __shared__ float sW[8];
  if ((tid & 31) == 0) sW[tid >> 5] = t;
  __syncthreads();
  if (tid == 0) {
    float s = 0.f;
    for (int w = 0; w < 8; ++w) s += sW[w];
    clsp[b * K2BPB + blockIdx.x] = s;
  }
}

// ---------------------------------------------------------------------------
// Kernel 3: finalize. One wave per batch reduces partials; thread 0 averages.
// ---------------------------------------------------------------------------
__global__ __launch_bounds__(256) void finalize_kernel(
    const float* __restrict__ clsp, const float* __restrict__ nposp,
    const float* __restrict__ regp, const float* __restrict__ ann,
    float* __restrict__ out)
{
  __shared__ float sC[BB], sR[BB];
  const int tid = threadIdx.x;
  const int w = tid >> 5, ln = tid & 31;
  if (w < BB) {
    float cs = 0.f;
    for (int i = ln; i < K2BPB; i += 32) cs += clsp[w * K2BPB + i];
    float np = 0.f, rg = 0.f;
    for (int i = ln; i < NBLK1; i += 32) {
      np += nposp[w * NBLK1 + i];
      rg += regp [w * NBLK1 + i];
    }
    int va = 0;
    for (int m = ln; m < MM; m += 32)
      va |= (ann[((size_t)w * MM + m) * 5 + 4] != -1.0f) ? 1 : 0;
    for (int off = 16; off; off >>= 1) {
      cs += __shfl_xor(cs, off, 32);
      np += __shfl_xor(np, off, 32);
      rg += __shfl_xor(rg, off, 32);
      va |= __shfl_xor(va, off, 32);
    }
    if (ln == 0) {
      float denom = fmaxf(np, 1.0f);
      sC[w] = va ? (cs / denom) : 0.0f;
      sR[w] = (np > 0.0f) ? rg / (4.0f * denom) : 0.0f;
    }
  }
  __syncthreads();
  if (tid == 0) {
    float c = 0.f, r = 0.f;
    for (int i = 0; i < BB; ++i) { c += sC[i]; r += sR[i]; }
    out[0] = c / (float)BB;
    out[1] = r / (float)BB;
  }
}

extern "C" void kernel_launch(void* const* d_in, const int* in_sizes, int n_in,
                              void* d_out, int out_size, void* d_ws, size_t ws_size,
                              hipStream_t stream) {
  const float* cls  = (const float*)d_in[0];  // (B,A,C) f32
  const float* regr = (const float*)d_in[1];  // (B,A,4) f32
  const float* anch = (const float*)d_in[2];  // (1,A,4) f32
  const float* ann  = (const float*)d_in[3];  // (B,M,5) f32
  (void)in_sizes; (void)n_in; (void)out_size; (void)ws_size;

  // Workspace layout (all slots rewritten every call -> graph-replay safe).
  int*   code  = (int*)d_ws;                                        // B*A
  float* clsp  = (float*)((char*)d_ws + (size_t)BB * AA * sizeof(int));
  float* nposp = clsp  + (size_t)BB * K2BPB;
  float* regp  = nposp + (size_t)BB * NBLK1;

  assign_kernel <<<dim3(NBLK1, BB), K1T, 0, stream>>>(anch, ann, regr, code, nposp, regp);
  clsloss_kernel<<<dim3(K2BPB, BB), 256, 0, stream>>>(cls, code, clsp);
  finalize_kernel<<<1, 256, 0, stream>>>(clsp, nposp, regp, ann, (float*)d_out);
}